// DecodePredictions_39857296507139
// MI455X (gfx1250) — compile-verified
//
#include <hip/hip_runtime.h>
#include <stdint.h>

#define BATCH       8
#define NANCH       76725
#define NPAD        76800        // multiple of 1024; 300 KB of f32 (<= 320 KB/WGP)
#define NCLS        80
#define CSTRIDE     84           // 4 box + 80 classes (336 B, 16-B aligned records)
#define MAXDET      100
#define NMS_THREADS 1024
#define DEC_THREADS 256

typedef uint32_t v4u __attribute__((ext_vector_type(4)));
typedef uint32_t v8u __attribute__((ext_vector_type(8)));

// ---------------------------------------------------------------------------
// Kernel 1: per-anchor decode. HBM-bound phase (206 MB in @ 23.3 TB/s ~ 9us).
// 2D grid: x covers NPAD anchors (exactly 300 blocks), y = batch -> no int div.
// Each thread owns one 336-byte record; loaded as 21 x b128.
// sigmoid(max(logits)) == max(sigmoid(logits)); argmax unchanged by sigmoid.
// ---------------------------------------------------------------------------
__global__ __launch_bounds__(DEC_THREADS)
void decode_kernel(const float* __restrict__ pred,
                   const float* __restrict__ anchors,
                   float* __restrict__ boxes,
                   float* __restrict__ corners,
                   float* __restrict__ cls,
                   float* __restrict__ scores)
{
    const int i = blockIdx.x * DEC_THREADS + threadIdx.x;   // [0, NPAD)
    const int b = blockIdx.y;                               // [0, BATCH)
    const size_t o = (size_t)b * NPAD + i;

    if (i >= NANCH) {
        // padding lanes: never selectable, never suppress anything
        scores[o] = -1.0f;
        cls[o]    = -1.0f;
        float4* c4 = (float4*)corners + o;
        *c4 = make_float4(-3.0e30f, -3.0e30f, -3.0e30f, -3.0e30f);
        float4* b4 = (float4*)boxes + o;
        *b4 = make_float4(0.0f, 0.0f, 0.0f, 0.0f);
        return;
    }

    const float4* q = (const float4*)(pred + ((size_t)b * NANCH + i) * CSTRIDE);
    __builtin_prefetch(q, 0, 1);                 // -> global_prefetch_b8

    float4 bp = q[0];                            // box regression

    // argmax over 80 logits as 20 x float4; first-max tie semantics (strict >)
    float maxv = -3.4e38f;
    int   maxi = 0;
    #pragma unroll 5
    for (int c = 0; c < NCLS / 4; ++c) {
        float4 v = q[1 + c];
        int base = 4 * c;
        if (v.x > maxv) { maxv = v.x; maxi = base + 0; }
        if (v.y > maxv) { maxv = v.y; maxi = base + 1; }
        if (v.z > maxv) { maxv = v.z; maxi = base + 2; }
        if (v.w > maxv) { maxv = v.w; maxi = base + 3; }
    }
    float conf = 1.0f / (1.0f + __expf(-maxv));

    float4 a = ((const float4*)anchors)[i];      // cx, cy, w, h

    float bx = bp.x * 0.1f, by = bp.y * 0.1f;    // BOX_VARIANCE
    float bw = bp.z * 0.2f, bh = bp.w * 0.2f;
    float cx = bx * a.z + a.x;
    float cy = by * a.w + a.y;
    float w  = __expf(bw) * a.z;
    float h  = __expf(bh) * a.w;

    float clsf = (float)maxi;
    float off  = clsf * 10000.0f;                // CLASS_OFFSET

    ((float4*)boxes)[o]   = make_float4(cx, cy, w, h);
    ((float4*)corners)[o] = make_float4(cx - 0.5f * w + off,
                                        cy - 0.5f * h + off,
                                        cx + 0.5f * w + off,
                                        cy + 0.5f * h + off);
    cls[o]    = clsf;
    scores[o] = (conf > 0.05f) ? conf : -1.0f;
}

// ---------------------------------------------------------------------------
// Kernel 2: one workgroup per batch image. Scores live in 300 KB of LDS
// (CDNA5: 320 KB/WGP), staged via TDM tensor_load_to_lds + per-lane
// global_load_async_to_lds_b128 (identical bytes; both gfx1250 async paths).
// The 100 sequential NMS steps then run out of LDS (scores, b128 reads) and
// L2-resident corners (9.8 MB total << 192 MB L2), skipping dead entries.
// ---------------------------------------------------------------------------
__global__ __launch_bounds__(NMS_THREADS)
void nms_kernel(const float* __restrict__ boxes,
                const float* __restrict__ corners,
                const float* __restrict__ cls,
                const float* __restrict__ scores_g,
                float* __restrict__ out)
{
    extern __shared__ float s_scores[];          // NPAD floats (300 KB)
    __shared__ float s_vals[NMS_THREADS / 32];
    __shared__ int   s_idx[NMS_THREADS / 32];
    __shared__ float s_sel[6];                   // x1,y1,x2,y2,valid

    const int b   = blockIdx.x;
    const int tid = threadIdx.x;
    const float*  sg = scores_g + (size_t)b * NPAD;
    const float4* c4 = (const float4*)corners + (size_t)b * NPAD;

    uint32_t lds_base = (uint32_t)(uintptr_t)(void*)s_scores;

    // ---- TDM: one descriptor DMA covering all NPAD floats (2D 7680 x 10) ----
    if (tid == 0) {
        uint64_t ga = (uint64_t)(uintptr_t)sg;
        v4u g0; v8u g1;
        g0[0] = 1u;                                   // count=1 valid descriptor
        g0[1] = lds_base;                             // lds_addr
        g0[2] = (uint32_t)ga;                         // global_addr[31:0]
        g0[3] = (uint32_t)(ga >> 32) | 0x80000000u;   // global_addr[56:32] | type=2
        g1[0] = 2u << 16;                             // data_size = 4 bytes
        g1[1] = (7680u & 0xFFFFu) << 16;              // tensor_dim0 lo16 @ bits 63:48
        g1[2] = 10u << 16;                            // tensor_dim0 hi=0 | tensor_dim1=10
        g1[3] = 7680u << 16;                          // tensor_dim1 hi=0 | tile_dim0=7680
        g1[4] = 10u;                                  // tile_dim1=10, tile_dim2=0
        g1[5] = 7680u;                                // tensor_dim0_stride[31:0]
        g1[6] = 0u;
        g1[7] = 0u;
        asm volatile("tensor_load_to_lds %0, %1" :: "s"(g0), "s"(g1) : "memory");
    }
    __builtin_amdgcn_s_wait_tensorcnt(0);

    // ---- per-lane async global -> LDS copy (same bytes) -----
    for (int i = tid * 4; i < NPAD; i += NMS_THREADS * 4) {
        uint32_t ldsa = lds_base + (uint32_t)i * 4u;
        uint32_t goff = (uint32_t)i * 4u;
        asm volatile("global_load_async_to_lds_b128 %0, %1, %2"
                     :: "v"(ldsa), "v"(goff), "s"(sg) : "memory");
    }
    asm volatile("s_wait_asynccnt 0" ::: "memory");
    __syncthreads();

    const float4* s4 = (const float4*)s_scores;  // ds_load_b128 scans

    // ---- 100 sequential NMS steps ----
    for (int it = 0; it < MAXDET; ++it) {
        // block-wide argmax (first-index on ties); vectorized LDS reads
        float bestv = -3.4e38f;
        int   besti = 0x7FFFFFFF;
        for (int i4 = tid; i4 < NPAD / 4; i4 += NMS_THREADS) {
            float4 s = s4[i4];
            int base = i4 * 4;
            if (s.x > bestv) { bestv = s.x; besti = base + 0; }
            if (s.y > bestv) { bestv = s.y; besti = base + 1; }
            if (s.z > bestv) { bestv = s.z; besti = base + 2; }
            if (s.w > bestv) { bestv = s.w; besti = base + 3; }
        }
        #pragma unroll
        for (int m = 16; m >= 1; m >>= 1) {      // wave32 reduction
            float ov = __shfl_xor(bestv, m, 32);
            int   oi = __shfl_xor(besti, m, 32);
            if (ov > bestv || (ov == bestv && oi < besti)) { bestv = ov; besti = oi; }
        }
        if ((tid & 31) == 0) { s_vals[tid >> 5] = bestv; s_idx[tid >> 5] = besti; }
        __syncthreads();

        if (tid == 0) {
            float bv = s_vals[0]; int bi = s_idx[0];
            for (int w = 1; w < NMS_THREADS / 32; ++w) {
                float v = s_vals[w]; int ix = s_idx[w];
                if (v > bv || (v == bv && ix < bi)) { bv = v; bi = ix; }
            }
            bool valid = bv > 0.0f;
            size_t base = (size_t)b * NPAD + bi;
            float* o = out + ((size_t)b * MAXDET + it) * 6;
            if (valid) {
                float4 bb = ((const float4*)boxes)[base];
                o[0] = bb.x; o[1] = bb.y; o[2] = bb.z; o[3] = bb.w;
                o[4] = cls[base]; o[5] = bv;
            } else {
                o[0] = 0.0f; o[1] = 0.0f; o[2] = 0.0f; o[3] = 0.0f;
                o[4] = -1.0f; o[5] = -1.0f;
            }
            float4 cc = c4[bi];
            s_sel[0] = cc.x; s_sel[1] = cc.y; s_sel[2] = cc.z; s_sel[3] = cc.w;
            s_sel[4] = valid ? 1.0f : 0.0f;
            s_scores[bi] = -1.0f;                // always clear winner
        }
        __syncthreads();

        if (s_sel[4] != 0.0f) {                  // suppress only when valid
            const float x1 = s_sel[0], y1 = s_sel[1], x2 = s_sel[2], y2 = s_sel[3];
            const float a1 = (x2 - x1) * (y2 - y1);
            for (int i = tid; i < NPAD; i += NMS_THREADS) {
                if (s_scores[i] <= 0.0f) continue;   // dead: skip L2 read (no-op in ref)
                float4 cb = c4[i];
                float lx = fmaxf(x1, cb.x), ly = fmaxf(y1, cb.y);
                float rx = fminf(x2, cb.z), ry = fminf(y2, cb.w);
                float iw = fmaxf(rx - lx, 0.0f), ih = fmaxf(ry - ly, 0.0f);
                float inter = iw * ih;
                float a2 = (cb.z - cb.x) * (cb.w - cb.y);
                float iou = inter / (a1 + a2 - inter + 1e-8f);
                if (iou > 0.5f) s_scores[i] = -1.0f;
            }
        }
        __syncthreads();
    }
}

// ---------------------------------------------------------------------------
extern "C" void kernel_launch(void* const* d_in, const int* in_sizes, int n_in,
                              void* d_out, int out_size, void* d_ws, size_t ws_size,
                              hipStream_t stream) {
    (void)in_sizes; (void)n_in; (void)out_size; (void)ws_size;
    // inputs: [0]=images (unused), [1]=predictions, [2]=anchor_boxes
    const float* pred    = (const float*)d_in[1];
    const float* anchors = (const float*)d_in[2];

    float* ws      = (float*)d_ws;                       // ~23.4 MB used
    float* boxes   = ws;                                 // B*NPAD*4
    float* corners = boxes   + (size_t)BATCH * NPAD * 4; // B*NPAD*4
    float* cls     = corners + (size_t)BATCH * NPAD * 4; // B*NPAD
    float* scores  = cls     + (size_t)BATCH * NPAD;     // B*NPAD
    float* out     = (float*)d_out;                      // B*100*6

    dim3 dgrid(NPAD / DEC_THREADS, BATCH, 1);            // 300 x 8, no divides
    decode_kernel<<<dgrid, DEC_THREADS, 0, stream>>>(
        pred, anchors, boxes, corners, cls, scores);
    nms_kernel<<<BATCH, NMS_THREADS, NPAD * sizeof(float), stream>>>(
        boxes, corners, cls, scores, out);
}